// BertForOrdering_75711683494267
// MI455X (gfx1250) — compile-verified
//
#include <hip/hip_runtime.h>
#include <hip/hip_bf16.h>
#include <math.h>

typedef __attribute__((ext_vector_type(2))) float v2f;
typedef __attribute__((ext_vector_type(8))) float v8f;

#define GM 2048   // B*N rows
#define GK 768    // H
#define GN 768    // H
#define NB 16     // batch
#define NT 128    // seq len
#define TM 8      // q-rows per score block
#define NEGV (-1e9f)

// Native gfx1250 TRANS op. Prefer the builtin when declared; otherwise emit
// v_tanh_f32 directly (non-volatile asm so the scheduler can interleave it
// with the VALU fma stream).
__device__ __forceinline__ float fast_tanh(float x) {
#if __has_builtin(__builtin_amdgcn_tanhf)
  return __builtin_amdgcn_tanhf(x);
#elif __has_builtin(__builtin_amdgcn_tanh_f32)
  return __builtin_amdgcn_tanh_f32(x);
#else
  float r;
  asm("v_tanh_f32 %0, %1" : "=v"(r) : "v"(x));
  return r;
#endif
}

// ---------------------------------------------------------------------------
// Kernel 1: C[M,N] = A[M,K] * W[K,N] + bias[N]  via V_WMMA_F32_16X16X4_F32.
// 8 waves / block, one 16x16 C tile per wave. grid = (128*48)/8 = 768 blocks.
// ---------------------------------------------------------------------------
__global__ void __launch_bounds__(256)
gemm_bias_wmma(const float* __restrict__ A, const float* __restrict__ W,
               const float* __restrict__ bias, float* __restrict__ C) {
  const int wave = threadIdx.x >> 5;
  const int lane = threadIdx.x & 31;
  const int tile = blockIdx.x * 8 + wave;          // 0..6143
  const int tilesN = GN / 16;                      // 48
  const int tm = (tile / tilesN) * 16;
  const int tn = (tile % tilesN) * 16;

  const int hf = lane >> 4;                        // half-wave select
  const int lm = lane & 15;

  // A frag: lanes 0-15 -> K = k0,k0+1 ; lanes 16-31 -> K = k0+2,k0+3 (row = tm+lm)
  const float* ap = A + (size_t)(tm + lm) * GK + 2 * hf;
  // B frag: lanes 0-15 -> rows k0,k0+1 ; lanes 16-31 -> rows k0+2,k0+3 (col = tn+lm)
  const float* bp = W + (size_t)(2 * hf) * GN + tn + lm;

  v8f c = {0.f, 0.f, 0.f, 0.f, 0.f, 0.f, 0.f, 0.f};

#pragma unroll 4
  for (int k = 0; k < GK; k += 4) {
    v2f a, b;
    a.x = ap[0];
    a.y = ap[1];
    b.x = bp[0];
    b.y = bp[GN];
    c = __builtin_amdgcn_wmma_f32_16x16x4_f32(false, a, false, b,
                                              (short)0, c, false, false);
    ap += 4;
    bp += 4 * GN;
  }

  const float bv = bias[tn + lm];
  // C/D layout: VGPR r holds row (r + 8*hf), col = lm
  float* cp = C + (size_t)(tm + 8 * hf) * GN + tn + lm;
#pragma unroll
  for (int r = 0; r < 8; ++r) cp[(size_t)r * GN] = c[r] + bv;
}

// ---------------------------------------------------------------------------
// Kernel 2: scores[b,t,j] = sum_h wt[h]*tanh(Q[b,t,h]+K[b,j,h]) + bt
// One block per (b, 8 consecutive t rows): 8 q-rows staged in LDS (24 KB),
// wt held in registers (fixed per-lane h-stripe), each loaded k-value reused
// across the 8 rows -> 1 LDS read + 1/8 L2 read per tanh, 8 independent
// accumulators to hide TRANS latency.
// ---------------------------------------------------------------------------
__global__ void __launch_bounds__(256)
scores_kernel(const float* __restrict__ Q, const float* __restrict__ Km,
              const float* __restrict__ wt, const float* __restrict__ bt,
              float* __restrict__ S) {
  __shared__ float qs[TM][GK];                 // 24 KB
  const int bt0 = blockIdx.x * TM;             // first (b*128+t) row of this block
  const int b = bt0 >> 7;                      // TM divides NT -> no batch crossing

  // stage 8 contiguous q rows
  const float* qbase = Q + (size_t)bt0 * GK;
  for (int i = threadIdx.x; i < TM * GK; i += 256)
    qs[i / GK][i % GK] = qbase[i];

  const int wave = threadIdx.x >> 5;
  const int lane = threadIdx.x & 31;

  // per-lane wt stripe into registers (h = lane + 32*i)
  float wreg[GK / 32];
#pragma unroll
  for (int i = 0; i < GK / 32; ++i) wreg[i] = wt[lane + 32 * i];

  __syncthreads();
  const float b0 = bt[0];

  for (int j = wave; j < NT; j += 8) {
    const float* kp = Km + ((size_t)(b * NT + j)) * GK + lane;
    float acc[TM];
#pragma unroll
    for (int t = 0; t < TM; ++t) acc[t] = 0.f;

#pragma unroll 4
    for (int i = 0; i < GK / 32; ++i) {
      const float kv = kp[32 * i];
      const int h = lane + 32 * i;
#pragma unroll
      for (int t = 0; t < TM; ++t)
        acc[t] = fmaf(wreg[i], fast_tanh(qs[t][h] + kv), acc[t]);
    }

#pragma unroll
    for (int t = 0; t < TM; ++t) {
      float a = acc[t];
#pragma unroll
      for (int off = 16; off > 0; off >>= 1) a += __shfl_down(a, off);
      if (lane == 0) S[(size_t)(bt0 + t) * NT + j] = a + b0;
    }
  }
}

// ---------------------------------------------------------------------------
// Kernel 3: masked row/column log-softmax losses. One block (128 thr) per
// batch; the whole 128x128 score slice lives in LDS (66 KB << 320 KB WGP LDS).
// pointed[t,j] <=> pos[j] < t where pos = inverse permutation of target.
// ---------------------------------------------------------------------------
__global__ void __launch_bounds__(128)
loss_kernel(const float* __restrict__ S, const int* __restrict__ target,
            const int* __restrict__ tgt_len, float* __restrict__ out) {
  __shared__ float sh[NT * 130];     // row stride 130 to dodge bank conflicts
  __shared__ int pos[NT];
  __shared__ float colmx[NT];
  __shared__ float colls[NT];
  __shared__ float red[NT];

  const int b = blockIdx.x;
  const int tid = threadIdx.x;       // doubles as row t and column j
  const int len = tgt_len[b];
  const int tg = target[b * NT + tid];
  pos[tg] = tid;

  for (int idx = tid; idx < NT * NT; idx += NT) {
    const int t = idx >> 7, j = idx & 127;
    sh[t * 130 + j] = S[((size_t)(b * NT + t)) * NT + j];
  }
  __syncthreads();

  // ---- row loss: log-softmax over j with pointed + column-validity mask ----
  const int t = tid;
  float mx = -3.402823e38f;
  for (int j = 0; j < NT; ++j) {
    const bool masked = (pos[j] < t) || (j >= len);
    const float e = masked ? NEGV : sh[t * 130 + j];
    mx = fmaxf(mx, e);
  }
  float sm = 0.f;
  for (int j = 0; j < NT; ++j) {
    const bool masked = (pos[j] < t) || (j >= len);
    const float e = masked ? NEGV : sh[t * 130 + j];
    sm += __expf(e - mx);
  }
  const bool tgMasked = (pos[tg] < t) || (tg >= len);
  const float eStar = tgMasked ? NEGV : sh[t * 130 + tg];
  float nll1 = -(eStar - mx - __logf(sm));
  if (t >= len) nll1 = 0.f;

  // ---- column loss: log-softmax over t (this thread owns column j=tid) ----
  const int j = tid;
  const bool jv = (j < len);
  float mx2 = -3.402823e38f;
  for (int r = 0; r < NT; ++r) {
    const float e = (jv && r < len) ? sh[r * 130 + j] : NEGV;
    mx2 = fmaxf(mx2, e);
  }
  float sm2 = 0.f;
  for (int r = 0; r < NT; ++r) {
    const float e = (jv && r < len) ? sh[r * 130 + j] : NEGV;
    sm2 += __expf(e - mx2);
  }
  colmx[j] = mx2;
  colls[j] = __logf(sm2);
  __syncthreads();

  const float e2 = (tg < len && t < len) ? sh[t * 130 + tg] : NEGV;
  float nll2 = -(e2 - colmx[tg] - colls[tg]);
  if (t >= len) nll2 = 0.f;

  // ---- combine with per-batch denominators, reduce, accumulate ----
  const float lf = (float)len;
  const float den1 = lf + 1e-20f - 1.0f;
  const float den2 = lf * den1;
  red[tid] = nll1 / den1 + nll2 / den2;
  __syncthreads();
  for (int s = 64; s > 0; s >>= 1) {
    if (tid < s) red[tid] += red[tid + s];
    __syncthreads();
  }
  if (tid == 0) atomicAdd(out, red[0] * (1.0f / (float)NB));
}

// ---------------------------------------------------------------------------
extern "C" void kernel_launch(void* const* d_in, const int* in_sizes, int n_in,
                              void* d_out, int out_size, void* d_ws, size_t ws_size,
                              hipStream_t stream) {
  (void)in_sizes; (void)n_in; (void)out_size; (void)ws_size;
  const float* dec  = (const float*)d_in[0];
  const float* sen  = (const float*)d_in[1];
  const float* Wq   = (const float*)d_in[2];
  const float* bq   = (const float*)d_in[3];
  const float* Wk   = (const float*)d_in[4];
  const float* bk   = (const float*)d_in[5];
  const float* wt   = (const float*)d_in[6];
  const float* btb  = (const float*)d_in[7];
  const int* target = (const int*)d_in[8];
  const int* tlen   = (const int*)d_in[9];

  float* ws = (float*)d_ws;
  float* Q  = ws;                                  // 2048*768 f32 (6 MB)
  float* Km = ws + (size_t)GM * GK;                // 2048*768 f32 (6 MB)
  float* S  = ws + 2 * (size_t)GM * GK;            // 16*128*128 f32 (1 MB)

  gemm_bias_wmma<<<768, 256, 0, stream>>>(dec, Wq, bq, Q);
  gemm_bias_wmma<<<768, 256, 0, stream>>>(sen, Wk, bk, Km);
  scores_kernel<<<(NB * NT) / TM, 256, 0, stream>>>(Q, Km, wt, btb, S);
  hipMemsetAsync(d_out, 0, sizeof(float), stream);
  loss_kernel<<<NB, NT, 0, stream>>>(S, target, tlen, (float*)d_out);
}